// PosEncodingHiENeRV_86036784874044
// MI455X (gfx1250) — compile-verified
//
#include <hip/hip_runtime.h>
#include <math.h>

typedef __attribute__((ext_vector_type(2))) float v2f;
typedef __attribute__((ext_vector_type(8))) float v8f;

#define FULL_T 60
#define FULL_H 270
#define FULL_W 480
#define NCH 14
#define NPATCH 64
#define PPAD_H 92
#define PPAD_W 98
#define EMBED_TOTAL (NPATCH * PPAD_H * PPAD_W * NCH)   // 8,078,336 floats
#define PI_F 3.14159265358979323846f

__device__ __forceinline__ int iclamp(int v, int lo, int hi) {
  return v < lo ? lo : (v > hi ? hi : v);
}

// Bilinear interpolation in (t,h) at grid-w index gw, channel c.
// Half-pixel centers, edge-clamped indices (== torch align_corners=False).
__device__ __forceinline__ float interp_th(const float* __restrict__ g,
                                           int Tg, int Hg, int Wg, int Cg,
                                           float tscale, float hscale,
                                           int t_pix, int h_pix, int gw, int c) {
  float st = ((float)t_pix + 0.5f) * tscale - 0.5f;
  float sh = ((float)h_pix + 0.5f) * hscale - 0.5f;
  float t0f = floorf(st), h0f = floorf(sh);
  float ft = st - t0f, fh = sh - h0f;
  int t0 = (int)t0f, h0 = (int)h0f;
  int t0c = iclamp(t0, 0, Tg - 1);
  int t1c = iclamp(t0 + 1, 0, Tg - 1);
  int h0c = iclamp(h0, 0, Hg - 1);
  int h1c = iclamp(h0 + 1, 0, Hg - 1);
  float v00 = g[((t0c * Hg + h0c) * Wg + gw) * Cg + c];
  float v01 = g[((t0c * Hg + h1c) * Wg + gw) * Cg + c];
  float v10 = g[((t1c * Hg + h0c) * Wg + gw) * Cg + c];
  float v11 = g[((t1c * Hg + h1c) * Wg + gw) * Cg + c];
  float lo = v00 + fh * (v01 - v00);
  float hi = v10 + fh * (v11 - v10);
  return lo + ft * (hi - lo);
}

// One block per (patch n, padded row oh). 256 threads = 8 waves.
//  - 168 threads stage the (t,h)-interpolated grid rows (all 3 levels) in LDS
//  - waves 0..6: one 16-wide w-tile each; 16(w) x 16(ch) tile via 3 chained
//    V_WMMA_F32_16X16X4_F32 (levels occupy disjoint channel columns)
//  - wave 7: t_manip (only on row 0)
__global__ void __launch_bounds__(256)
posenc_hienerv_kernel(const int* __restrict__ idx,
                      const float* __restrict__ g0,
                      const float* __restrict__ g1,
                      const float* __restrict__ g2,
                      float* __restrict__ out) {
  __shared__ float srow[176];  // L0:[0,128) L1:[128,160) L2:[160,168)

  const int blk = blockIdx.x;
  const int n = blk / PPAD_H;
  const int oh = blk % PPAD_H;

  const int idx_t = idx[3 * n + 0];
  const int idx_h = idx[3 * n + 1];
  const int idx_w = idx[3 * n + 2];

  const int raw_h = idx_h * 90 + oh - 1;
  const float mh = (raw_h >= 0 && raw_h < FULL_H) ? 1.0f : 0.0f;
  const int h_pix = iclamp(raw_h, 0, FULL_H - 1);
  const int t_pix = idx_t;  // patch_t == 1, pad_t == 0: always in range

  const int tid = threadIdx.x;

  // ---- Stage (t,h)-interpolated rows into LDS (h-mask folded in) ----
  if (tid < 128) {                      // level 0: 16 gw x 8 ch
    int gw = tid >> 3, c = tid & 7;
    srow[tid] = mh * interp_th(g0, 60, 16, 16, 8, 1.0f, 16.0f / 270.0f,
                               t_pix, h_pix, gw, c);
  } else if (tid < 160) {               // level 1: 8 gw x 4 ch
    int q = tid - 128;
    int gw = q >> 2, c = q & 3;
    srow[128 + q] = mh * interp_th(g1, 30, 8, 8, 4, 0.5f, 8.0f / 270.0f,
                                   t_pix, h_pix, gw, c);
  } else if (tid < 168) {               // level 2: 4 gw x 2 ch
    int q = tid - 160;
    int gw = q >> 1, c = q & 1;
    srow[160 + q] = mh * interp_th(g2, 15, 4, 4, 2, 0.25f, 4.0f / 270.0f,
                                   t_pix, h_pix, gw, c);
  }
  __syncthreads();

  const int wave = tid >> 5;
  const int lane = tid & 31;
  const int m = lane & 15;       // A row / B-N (channel) column
  const int khalf = lane >> 4;   // which K-pair this half-wave holds

  if (wave < 7) {
    const int w0 = wave * 16;
    const int wout = w0 + m;
    const int raw_w = idx_w * 96 + wout - 1;
    const float mwf = (raw_w >= 0 && raw_w < FULL_W && wout < PPAD_W) ? 1.0f : 0.0f;
    const int w_pix = iclamp(raw_w, 0, FULL_W - 1);

    v8f acc = {};

    // One level: build A (w-interp weights, 16x4) and B (4 grid-w entries x
    // 16 channel columns, level channels at colOff..colOff+Cg), one WMMA.
    // Fully branchless so EXEC stays all-1s through the WMMA sequence.
    auto do_level = [&](int ldsOff, int Wg, int Cg, int colOff, float wscale) {
      float srcw = ((float)w_pix + 0.5f) * wscale - 0.5f;
      float i0f = floorf(srcw);
      float fw = srcw - i0f;
      int i0 = (int)i0f;
      int i0c = iclamp(i0, 0, Wg - 1);
      int i1c = iclamp(i0 + 1, 0, Wg - 1);
      // Wave-uniform base grid-w index: lane 0 holds m==0 (smallest w).
      int wbase = __builtin_amdgcn_readfirstlane(i0c);
      float a[4];
#pragma unroll
      for (int k = 0; k < 4; ++k) {
        float w = (i0c - wbase == k) ? (1.0f - fw) : 0.0f;
        w += (i1c - wbase == k) ? fw : 0.0f;
        a[k] = w * mwf;
      }
      // A layout (16x4 f32): lanes 0-15 -> K=0,1 ; lanes 16-31 -> K=2,3
      v2f A;
      A.x = khalf ? a[2] : a[0];
      A.y = khalf ? a[3] : a[1];
      // B layout (4x16 f32): lanes 0-15 -> K=0,1 ; lanes 16-31 -> K=2,3;
      // column (N) = lane&15. Columns outside this level's channel range are
      // zeroed by mask (load a clamped duplicate address unconditionally).
      int cl = m - colOff;
      float cmask = ((cl >= 0) && (cl < Cg)) ? 1.0f : 0.0f;
      int clc = iclamp(cl, 0, Cg - 1);
      int gA = iclamp(wbase + 2 * khalf + 0, 0, Wg - 1);
      int gB = iclamp(wbase + 2 * khalf + 1, 0, Wg - 1);
      v2f B;
      B.x = cmask * srow[ldsOff + gA * Cg + clc];
      B.y = cmask * srow[ldsOff + gB * Cg + clc];
      acc = __builtin_amdgcn_wmma_f32_16x16x4_f32(
          /*neg_a=*/false, A, /*neg_b=*/false, B,
          /*c_mod=*/(short)0, acc, /*reuse_a=*/false, /*reuse_b=*/false);
    };

    do_level(0, 16, 8, 0, 16.0f / 480.0f);    // level 0 -> channels 0..7
    do_level(128, 8, 4, 8, 8.0f / 480.0f);    // level 1 -> channels 8..11
    do_level(160, 4, 2, 12, 4.0f / 480.0f);   // level 2 -> channels 12..13

    // D layout: VGPR r: lanes 0-15 -> row M=r, lanes 16-31 -> row M=r+8;
    // column N = lane&15 = channel.
    const int ch = m;
    float* obase = out + ((size_t)n * PPAD_H + oh) * (PPAD_W * NCH);
#pragma unroll
    for (int r = 0; r < 8; ++r) {
      int wm = w0 + r + 8 * khalf;
      if (ch < NCH && wm < PPAD_W) obase[wm * NCH + ch] = acc[r];
    }
  } else if (oh == 0 && lane < 16) {
    // t_manip: [N,16] = [sin(idx_t * 2^j * pi) j=0..7, cos(...) j=0..7]
    // Hardware trig (v_sin/v_cos) — tiny side output, not precision-critical.
    int j = lane & 7;
    float v = (float)idx_t * exp2f((float)j) * PI_F;
    out[EMBED_TOTAL + n * 16 + lane] = (lane < 8) ? __sinf(v) : __cosf(v);
  }
}

extern "C" void kernel_launch(void* const* d_in, const int* in_sizes, int n_in,
                              void* d_out, int out_size, void* d_ws, size_t ws_size,
                              hipStream_t stream) {
  (void)in_sizes; (void)n_in; (void)d_ws; (void)ws_size; (void)out_size;
  const int* idx = (const int*)d_in[0];
  // d_in[1]=idx_max, d_in[2]=padding are compile-time constants here.
  const float* g0 = (const float*)d_in[3];
  const float* g1 = (const float*)d_in[4];
  const float* g2 = (const float*)d_in[5];
  float* out = (float*)d_out;

  dim3 grid(NPATCH * PPAD_H);  // 5888 blocks: one per (patch, padded row)
  dim3 block(256);             // 8 wave32: 7 WMMA tiles + 1 manip wave
  posenc_hienerv_kernel<<<grid, block, 0, stream>>>(idx, g0, g1, g2, out);
}